// ModulatedDeformConv_2534030705072
// MI455X (gfx1250) — compile-verified
//
#include <hip/hip_runtime.h>
#include <hip/hip_bf16.h>

typedef __attribute__((ext_vector_type(2))) float v2f;
typedef __attribute__((ext_vector_type(8))) float v8f;

#define DH   128
#define DW   128
#define DHW  (DH * DW)
#define NCIN 64
#define NCOUT 64
#define NK   9

// One workgroup (8 waves) = one (n, ho) output row of 128 pixels.
// Each wave: 16 pixels x 64 Cout, 4 fp32 WMMA accumulators.
// Reduction order: kred = k*64 + c (tap-major), chunks of 4 for 16x16x4 f32 WMMA.
__global__ __launch_bounds__(256, 1) void dcnv2_wmma_kernel(
    const float* __restrict__ x, const float* __restrict__ offs,
    const float* __restrict__ mask, const float* __restrict__ weight,
    const float* __restrict__ bias, float* __restrict__ out)
{
    // Weight tap slice, pre-swizzled to WMMA A layout: wl[c4][half][cout][j] (float2 per A operand)
    __shared__ float wLds[4096];          // 16 KB
    // Per-wave cols tile in WMMA B layout: cl[c4][half][pixel][j]
    __shared__ float colsLds[8][1024];    // 32 KB

    const int tid  = threadIdx.x;
    const int wave = tid >> 5;
    const int lane = tid & 31;
    const int p    = lane & 15;   // pixel within the wave's 16-pixel group (WMMA N)
    const int half = lane >> 4;   // K-half selector in A/B layouts

    const int wg = blockIdx.x;    // 0..511
    const int n  = wg >> 7;
    const int ho = wg & 127;
    const int wo = wave * 16 + p;

    v8f acc0 = {}, acc1 = {}, acc2 = {}, acc3 = {};

    const float* xn    = x + (size_t)n * NCIN * DHW;
    const int    hw    = ho * DW + wo;
    const float* offp  = offs + (size_t)n * 2 * NK * DHW + hw;
    const float* maskp = mask + (size_t)n * NK * DHW + hw;
    float*       cl    = colsLds[wave];

    for (int k = 0; k < NK; ++k) {
        __syncthreads();  // protect LDS reuse across taps

        // ---- Stage weight slice for tap k, reordered into WMMA A layout ----
        // flat idx = ((c4*2+half)*64 + cout)*2 + j ; channel c = c4*4 + half*2 + j
        #pragma unroll
        for (int t = 0; t < 16; ++t) {
            int idx = tid + t * 256;
            int c4  = idx >> 8;
            int hf  = (idx >> 7) & 1;
            int co  = (idx >> 1) & 63;
            int j   = idx & 1;
            int c   = c4 * 4 + hf * 2 + j;
            wLds[idx] = weight[co * (NCIN * NK) + c * NK + k];
        }

        // ---- Deformable bilinear sampling for this tap (per lane: pixel p) ----
        const int ky = k / 3, kx = k % 3;
        float offy = offp[(2 * k + 0) * DHW];
        float offx = offp[(2 * k + 1) * DHW];
        float mval = maskp[k * DHW];

        float py  = (float)(ho - 1 + ky) + offy;   // STRIDE=1, PAD=1, DIL=1
        float px  = (float)(wo - 1 + kx) + offx;
        float y0f = floorf(py), x0f = floorf(px);
        float ly  = py - y0f,   lx  = px - x0f;
        int   y0  = (int)y0f,   x0  = (int)x0f;
        int   y1  = y0 + 1,     x1  = x0 + 1;
        float wy0 = 1.0f - ly,  wx0 = 1.0f - lx;

        bool vy0 = (y0 >= 0) && (y0 < DH);
        bool vy1 = (y1 >= 0) && (y1 < DH);
        bool vx0 = (x0 >= 0) && (x0 < DW);
        bool vx1 = (x1 >= 0) && (x1 < DW);
        // zero-pad via weights, clamp addresses -> branch-free gathers
        float w00 = (vy0 && vx0) ? wy0 * wx0 * mval : 0.0f;
        float w01 = (vy0 && vx1) ? wy0 * lx  * mval : 0.0f;
        float w10 = (vy1 && vx0) ? ly  * wx0 * mval : 0.0f;
        float w11 = (vy1 && vx1) ? ly  * lx  * mval : 0.0f;

        int yc0 = min(max(y0, 0), DH - 1);
        int yc1 = min(max(y1, 0), DH - 1);
        int xc0 = min(max(x0, 0), DW - 1);
        int xc1 = min(max(x1, 0), DW - 1);
        int i00 = yc0 * DW + xc0;
        int i01 = yc0 * DW + xc1;
        int i10 = yc1 * DW + xc0;
        int i11 = yc1 * DW + xc1;

        // Each lane produces 32 channels (its half), written as float2 pairs in B layout.
        #pragma unroll
        for (int c4 = 0; c4 < 16; ++c4) {
            v2f pair;
            #pragma unroll
            for (int j = 0; j < 2; ++j) {
                int c = c4 * 4 + half * 2 + j;
                const float* xc = xn + c * DHW;
                pair[j] = w00 * xc[i00] + w01 * xc[i01] +
                          w10 * xc[i10] + w11 * xc[i11];
            }
            *(v2f*)&cl[c4 * 64 + half * 32 + p * 2] = pair;
        }

        __syncthreads();  // wLds + colsLds visible to all lanes

        // ---- fp32 WMMA: 16 K-chunks of 4, 4 Cout tiles of 16 ----
        #pragma unroll
        for (int c4 = 0; c4 < 16; ++c4) {
            v2f B = *(const v2f*)&cl[c4 * 64 + half * 32 + p * 2];
            const float* wb = &wLds[c4 * 256 + half * 128];
            v2f A0 = *(const v2f*)&wb[(0 * 16 + p) * 2];
            v2f A1 = *(const v2f*)&wb[(1 * 16 + p) * 2];
            v2f A2 = *(const v2f*)&wb[(2 * 16 + p) * 2];
            v2f A3 = *(const v2f*)&wb[(3 * 16 + p) * 2];
            acc0 = __builtin_amdgcn_wmma_f32_16x16x4_f32(false, A0, false, B, (short)0, acc0, false, false);
            acc1 = __builtin_amdgcn_wmma_f32_16x16x4_f32(false, A1, false, B, (short)0, acc1, false, false);
            acc2 = __builtin_amdgcn_wmma_f32_16x16x4_f32(false, A2, false, B, (short)0, acc2, false, false);
            acc3 = __builtin_amdgcn_wmma_f32_16x16x4_f32(false, A3, false, B, (short)0, acc3, false, false);
        }
    }

    // ---- Write out: lane l, vgpr v -> M = v + 8*half, N = p ----
    float* outb = out + (size_t)n * NCOUT * DHW + ho * DW + wo;
    #pragma unroll
    for (int v = 0; v < 8; ++v) {
        int m0 = v + 8 * half;
        outb[(0  + m0) * DHW] = acc0[v] + bias[0  + m0];
        outb[(16 + m0) * DHW] = acc1[v] + bias[16 + m0];
        outb[(32 + m0) * DHW] = acc2[v] + bias[32 + m0];
        outb[(48 + m0) * DHW] = acc3[v] + bias[48 + m0];
    }
}

extern "C" void kernel_launch(void* const* d_in, const int* in_sizes, int n_in,
                              void* d_out, int out_size, void* d_ws, size_t ws_size,
                              hipStream_t stream) {
    (void)in_sizes; (void)n_in; (void)out_size; (void)d_ws; (void)ws_size;
    const float* x      = (const float*)d_in[0];
    const float* offset = (const float*)d_in[1];
    const float* mask   = (const float*)d_in[2];
    const float* weight = (const float*)d_in[3];
    const float* bias   = (const float*)d_in[4];
    float* out = (float*)d_out;

    dcnv2_wmma_kernel<<<dim3(4 * 128), dim3(256), 0, stream>>>(
        x, offset, mask, weight, bias, out);
}